// DWTConv_39857296507523
// MI455X (gfx1250) — compile-verified
//
#include <hip/hip_runtime.h>
#include <stdint.h>

// ---------------------------------------------------------------------------
// Haar 2x2 DWT, fp32. Memory-bound: 1 GiB in + 1 GiB out, 0.5 FLOP/byte ->
// ~92 us floor at 23.3 TB/s. Data path: gfx1250 async global->LDS (ASYNCcnt)
// pipeline, statically double-buffered (unroll-2, no runtime buffer select),
// ds_load_b128 reads, four full-width non-temporal b128 stores per thread.
// WMMA is deliberately not used: the 4x4 butterfly on v_wmma_f32_16x16x4_f32
// would waste >=75% of the MACs and force half-wave 64B stores on a kernel
// whose only limit is the store/load pipe.
// ---------------------------------------------------------------------------

typedef __attribute__((ext_vector_type(4))) float v4f;

#define TILE_FLOATS 4096       // one tile = 8 input rows x 512 cols (16 KB, contiguous)
#define K_TILES     8          // tiles per block
#define NBLOCKS     8192       // 8192 * 8 = 65536 tiles = whole tensor
#define P_PLANE     67108864LL // floats per output plane (8*128*256*256)

__device__ __forceinline__ void async_tile_copy(uint32_t lds_addr, const float* g)
{
    // 64 B per lane: four b128 async copies. The instruction offset applies to
    // BOTH the LDS and global address (CDNA5 ISA), so one LDS
    // base VGPR + one global address pair covers all four.
    asm volatile("global_load_async_to_lds_b128 %0, %1, off"           :: "v"(lds_addr), "v"(g) : "memory");
    asm volatile("global_load_async_to_lds_b128 %0, %1, off offset:16" :: "v"(lds_addr), "v"(g) : "memory");
    asm volatile("global_load_async_to_lds_b128 %0, %1, off offset:32" :: "v"(lds_addr), "v"(g) : "memory");
    asm volatile("global_load_async_to_lds_b128 %0, %1, off offset:48" :: "v"(lds_addr), "v"(g) : "memory");
}

__device__ __forceinline__ void butterfly_store(const float* __restrict__ buf, int eix,
                                                float* __restrict__ o_hh, float* __restrict__ o_hl,
                                                float* __restrict__ o_lh, float* __restrict__ o_ll,
                                                long long ob)
{
    // Even input row: a0 b0 a1 b1 | a2 b2 a3 b3 ; odd row: c/d likewise.
    v4f e0 = *(const v4f*)(buf + eix);
    v4f e1 = *(const v4f*)(buf + eix + 4);
    v4f o0 = *(const v4f*)(buf + eix + 512);
    v4f o1 = *(const v4f*)(buf + eix + 516);

    const float a[4] = { e0.x, e0.z, e1.x, e1.z };
    const float b[4] = { e0.y, e0.w, e1.y, e1.w };
    const float c[4] = { o0.x, o0.z, o1.x, o1.z };
    const float d[4] = { o0.y, o0.w, o1.y, o1.w };

    v4f hh, hl, lh, ll;
#pragma unroll
    for (int k = 0; k < 4; ++k) {
        float sab = a[k] + b[k], dab = a[k] - b[k];
        float scd = c[k] + d[k], dcd = c[k] - d[k];
        hh[k] = 0.5f * (dab - dcd);   // a - b - c + d
        hl[k] = 0.5f * (sab - scd);   // a + b - c - d
        lh[k] = 0.5f * (dab + dcd);   // a - b + c - d
        ll[k] = 0.5f * (sab + scd);   // a + b + c + d
    }

    // Write-once streaming outputs -> non-temporal b128, wave-contiguous 512 B.
    __builtin_nontemporal_store(hh, (v4f*)(o_hh + ob));
    __builtin_nontemporal_store(hl, (v4f*)(o_hl + ob));
    __builtin_nontemporal_store(lh, (v4f*)(o_lh + ob));
    __builtin_nontemporal_store(ll, (v4f*)(o_ll + ob));
}

__global__ __launch_bounds__(256) void
dwt_haar_kernel(const float* __restrict__ x, float* __restrict__ out)
{
    __shared__ __align__(64) float lds[2][TILE_FLOATS];   // 32 KB double buffer

    const int       t  = threadIdx.x;
    const long long q0 = (long long)blockIdx.x * K_TILES;

    // This thread's 64 B slice of each tile.
    const float* g = x + q0 * TILE_FLOATS + 16 * t;

    const uint32_t lb0 = (uint32_t)(uintptr_t)&lds[0][16 * t];
    const uint32_t lb1 = (uint32_t)(uintptr_t)&lds[1][16 * t];

    async_tile_copy(lb0, g);   // prefetch tile 0

    // Thread t produces output elements j = 4t..4t+3 of the tile's 1024
    // outputs per plane (4 output rows x 256 cols).
    const int eix = (t >> 6) * 1024 + 8 * (t & 63);

    float* __restrict__ o_hh = out;
    float* __restrict__ o_hl = out +     P_PLANE;
    float* __restrict__ o_lh = out + 2 * P_PLANE;
    float* __restrict__ o_ll = out + 3 * P_PLANE;

    long long ob = q0 * 1024 + 4 * t;

#pragma unroll 1
    for (int i = 0; i < K_TILES; i += 2) {
        // ---- phase A: tile i lives in lds[0] ----
        async_tile_copy(lb1, g + TILE_FLOATS);            // i+1 < K always here
        asm volatile("s_wait_asynccnt 0x4" ::: "memory"); // tile i landed
        __syncthreads();
        butterfly_store(lds[0], eix, o_hh, o_hl, o_lh, o_ll, ob);
        __syncthreads();                                  // lds[0] free for i+2

        // ---- phase B: tile i+1 lives in lds[1] ----
        if (i + 2 < K_TILES) {
            async_tile_copy(lb0, g + 2 * TILE_FLOATS);
            asm volatile("s_wait_asynccnt 0x4" ::: "memory");
        } else {
            asm volatile("s_wait_asynccnt 0x0" ::: "memory");
        }
        __syncthreads();
        butterfly_store(lds[1], eix, o_hh, o_hl, o_lh, o_ll, ob + 1024);
        __syncthreads();                                  // lds[1] free for i+3

        g  += 2 * TILE_FLOATS;
        ob += 2048;
    }
}

extern "C" void kernel_launch(void* const* d_in, const int* in_sizes, int n_in,
                              void* d_out, int out_size, void* d_ws, size_t ws_size,
                              hipStream_t stream)
{
    (void)in_sizes; (void)n_in; (void)out_size; (void)d_ws; (void)ws_size;
    const float* x   = (const float*)d_in[0];
    float*       out = (float*)d_out;
    dwt_haar_kernel<<<NBLOCKS, 256, 0, stream>>>(x, out);
}